// Random_effects_64080912056938
// MI455X (gfx1250) — compile-verified
//
#include <hip/hip_runtime.h>
#include <cstdint>
#include <cstddef>

// ---------------------------------------------------------------------------
// Random effects broadcast:  out[b,t] = x[b,t] * w_slope[b,seg[t]] + w_inter[b,seg[t]]
//   w_inter[b,s] = sampler_inter[b,s] * exp(ranef_inter[b,s]) * (lengths[s] > 2)
//   w_slope[b,s] = sampler_slope[b,s] * exp(ranef_slope[b,s]) * (lengths[s] > 5)
//
// Memory-bound streaming kernel (~270 MB HBM traffic -> ~12 us @ 23.3 TB/s).
// CDNA5 paths used:
//   * async global->LDS DMA for the shared seg_ids tile
//     (global_load_async_to_lds_b128 + s_wait_asynccnt, ASYNCcnt-tracked)
//   * non-temporal (TH=NT) b128 loads/stores for the read-once x stream and
//     write-once out stream, keeping the 4 MB weight tables hot in the 192 MB
//     L2 for the per-measurement gathers
//   * wave32-sized 256-thread blocks (8 waves / block)
// ---------------------------------------------------------------------------

#define T_TILE  1024   // t-values per block (4 per thread, 256 threads)
#define THREADS 256

// Native vector type: required by __builtin_nontemporal_{load,store}
typedef float v4f __attribute__((ext_vector_type(4)));

// Low 32 bits of a generic pointer into LDS == wave-relative LDS byte offset
// (flat LDS aperture truncates to addr[31:0]).
__device__ __forceinline__ unsigned lds_offset(const void* p) {
    return (unsigned)(uintptr_t)p;
}

// ---------------- Phase 1: per-(b,s) weights (2D grid, no div/mod) ---------
__global__ __launch_bounds__(THREADS)
void re_weights_kernel(const float* __restrict__ ranef_i,
                       const float* __restrict__ ranef_s,
                       const float* __restrict__ samp_i,
                       const float* __restrict__ samp_s,
                       const int*   __restrict__ lengths,
                       float* __restrict__ w_i,
                       float* __restrict__ w_s,
                       int S) {
    const int s = blockIdx.x * blockDim.x + threadIdx.x;
    const int b = blockIdx.y;
    if (s >= S) return;
    const size_t i = (size_t)b * S + s;
    const int len = lengths[s];
    const float mi = (len > 2) ? 1.0f : 0.0f;
    const float ms = (len > 5) ? 1.0f : 0.0f;
    w_i[i] = samp_i[i] * expf(ranef_i[i]) * mi;
    w_s[i] = samp_s[i] * expf(ranef_s[i]) * ms;
}

// ---------------- Phase 2: streaming gather/broadcast ----------------------
__global__ __launch_bounds__(THREADS)
void re_gather_kernel(const float* __restrict__ x,
                      const int*   __restrict__ seg_ids,
                      const float* __restrict__ w_i,
                      const float* __restrict__ w_s,
                      float* __restrict__ out,
                      int T, int S) {
    __shared__ int s_sid[T_TILE];

    const int tid   = threadIdx.x;
    const int b     = blockIdx.y;
    const int t0    = blockIdx.x * T_TILE;       // tile base in t
    const int tbase = t0 + tid * 4;              // this thread's first t

    // ---- stage seg_ids tile into LDS via CDNA5 async DMA (ASYNCcnt) ----
    if (tbase + 4 <= T) {
        unsigned lds   = lds_offset(&s_sid[tid * 4]);
        unsigned goff  = (unsigned)(tid * 4) * 4u;        // byte offset in tile
        const int* gbase = seg_ids + t0;                   // uniform -> SGPR pair
        asm volatile("global_load_async_to_lds_b128 %0, %1, %2"
                     :
                     : "v"(lds), "v"(goff), "s"(gbase)
                     : "memory");
    } else {
        #pragma unroll
        for (int k = 0; k < 4; ++k) {
            int t = tbase + k;
            s_sid[tid * 4 + k] = (t < T) ? seg_ids[t] : 0;
        }
    }
    asm volatile("s_wait_asynccnt 0" ::: "memory");
    __syncthreads();

    const float* __restrict__ wib = w_i + (size_t)b * S;
    const float* __restrict__ wsb = w_s + (size_t)b * S;
    const float* __restrict__ xb  = x   + (size_t)b * T;
    float*       __restrict__ ob  = out + (size_t)b * T;

    if (tbase + 4 <= T) {
        // T is a multiple of 4 and tbase is a multiple of 4 -> 16B aligned.
        // NT hint: read-once stream, don't evict the L2-resident weight tables.
        v4f xv = __builtin_nontemporal_load((const v4f*)(xb + tbase));
        int i0 = s_sid[tid * 4 + 0];
        int i1 = s_sid[tid * 4 + 1];
        int i2 = s_sid[tid * 4 + 2];
        int i3 = s_sid[tid * 4 + 3];
        v4f o;
        o.x = fmaf(xv.x, wsb[i0], wib[i0]);
        o.y = fmaf(xv.y, wsb[i1], wib[i1]);
        o.z = fmaf(xv.z, wsb[i2], wib[i2]);
        o.w = fmaf(xv.w, wsb[i3], wib[i3]);
        __builtin_nontemporal_store(o, (v4f*)(ob + tbase));
    } else {
        #pragma unroll
        for (int k = 0; k < 4; ++k) {
            int t = tbase + k;
            if (t < T) {
                int sid = s_sid[tid * 4 + k];
                ob[t] = fmaf(xb[t], wsb[sid], wib[sid]);
            }
        }
    }
}

// ---------------- Fallback: fused single-pass (no workspace) ---------------
__global__ void re_fused_kernel(const float* __restrict__ x,
                                const int*   __restrict__ seg_ids,
                                const float* __restrict__ ranef_i,
                                const float* __restrict__ ranef_s,
                                const float* __restrict__ samp_i,
                                const float* __restrict__ samp_s,
                                const int*   __restrict__ lengths,
                                float* __restrict__ out,
                                int T, int S) {
    int t = blockIdx.x * blockDim.x + threadIdx.x;
    int b = blockIdx.y;
    if (t >= T) return;
    int sid  = seg_ids[t];
    int len  = lengths[sid];
    size_t w = (size_t)b * S + sid;
    float wi = samp_i[w] * expf(ranef_i[w]) * ((len > 2) ? 1.0f : 0.0f);
    float ws = samp_s[w] * expf(ranef_s[w]) * ((len > 5) ? 1.0f : 0.0f);
    size_t o = (size_t)b * T + t;
    out[o] = fmaf(x[o], ws, wi);
}

// ---------------------------------------------------------------------------
extern "C" void kernel_launch(void* const* d_in, const int* in_sizes, int n_in,
                              void* d_out, int out_size, void* d_ws, size_t ws_size,
                              hipStream_t stream) {
    const float* x       = (const float*)d_in[0];  // [B,T]
    const float* ranef_i = (const float*)d_in[1];  // [B,S]
    const float* ranef_s = (const float*)d_in[2];  // [B,S]
    const float* samp_i  = (const float*)d_in[3];  // [B,S]
    const float* samp_s  = (const float*)d_in[4];  // [B,S]
    const int*   seg_ids = (const int*)d_in[5];    // [T]
    const int*   lengths = (const int*)d_in[6];    // [S]

    const int S  = in_sizes[6];
    const int BS = in_sizes[1];
    const int T  = in_sizes[5];
    const int B  = BS / S;

    const size_t ws_need = (size_t)2 * (size_t)BS * sizeof(float);

    if (ws_size >= ws_need) {
        float* w_i = (float*)d_ws;
        float* w_s = w_i + BS;

        dim3 wgrid((S + THREADS - 1) / THREADS, B);
        re_weights_kernel<<<wgrid, THREADS, 0, stream>>>(
            ranef_i, ranef_s, samp_i, samp_s, lengths, w_i, w_s, S);

        dim3 grid((T + T_TILE - 1) / T_TILE, B);
        re_gather_kernel<<<grid, THREADS, 0, stream>>>(
            x, seg_ids, w_i, w_s, (float*)d_out, T, S);
    } else {
        dim3 grid((T + THREADS - 1) / THREADS, B);
        re_fused_kernel<<<grid, THREADS, 0, stream>>>(
            x, seg_ids, ranef_i, ranef_s, samp_i, samp_s, lengths,
            (float*)d_out, T, S);
    }
}